// GraphEncoder_41850161332368
// MI455X (gfx1250) — compile-verified
//
#include <hip/hip_runtime.h>

typedef float v2f __attribute__((ext_vector_type(2)));
typedef float v8f __attribute__((ext_vector_type(8)));

#define KC 64  // K-chunk staged in LDS per block

// ---------------------------------------------------------------------------
// Degree / normalization
// ---------------------------------------------------------------------------
__global__ void deg_init(float* __restrict__ deg, int n) {
  int i = blockIdx.x * blockDim.x + threadIdx.x;
  if (i < n) deg[i] = 1.0f;  // self loop
}

__global__ void deg_count(const int* __restrict__ dst, float* __restrict__ deg, int E) {
  int e = blockIdx.x * blockDim.x + threadIdx.x;
  if (e < E) unsafeAtomicAdd(&deg[dst[e]], 1.0f);
}

__global__ void deg_rsqrt(float* __restrict__ deg, int n) {
  int i = blockIdx.x * blockDim.x + threadIdx.x;
  if (i < n) deg[i] = rsqrtf(deg[i]);
}

__global__ void zero_f32(float* __restrict__ p, size_t n) {
  size_t i = (size_t)blockIdx.x * blockDim.x + threadIdx.x;
  if (i < n) p[i] = 0.0f;
}

// ---------------------------------------------------------------------------
// WMMA fp32 GEMM: C[M,N] = A[M,K] @ B[K,N] (+ bias). Row-major.
// Block = 256 threads (8 waves). blockIdx -> (mBlk, g): 256 rows x 64 cols.
// Wave w computes a 32x64 tile (2 row-halves x 4 col-tiles, 8 v8f accums).
// B chunk [KC x 64] double-buffered in LDS, stored as K-pairs so one
// fragment = one ds_load_b64:  Bs2[p][c] = {B[2p][c], B[2p+1][c]}.
// V_WMMA_F32_16X16X4_F32 fragment layout (wave32):
//   A: lanes 0-15 row=lane  K={0,1}; lanes 16-31 row=lane-16 K={2,3}
//   B: lanes 0-15 col=lane  K={0,1}; lanes 16-31 col=lane-16 K={2,3}
//   C/D: VGPR v, lane<16 -> (M=v,N=lane); lane>=16 -> (M=v+8,N=lane-16)
// Requires M%32==0 (100000=3125*32), N%64==0, K%KC==0.
// ---------------------------------------------------------------------------
__global__ __launch_bounds__(256) void gemm_wmma_f32(
    const float* __restrict__ A, const float* __restrict__ B,
    const float* __restrict__ bias, float* __restrict__ C,
    int M, int K, int N) {
  __shared__ float Bs[2][KC * 64];  // 2 x 16 KB

  const int G = N >> 6;
  const int g = blockIdx.x % G;
  const int mBlk = blockIdx.x / G;
  const int nBase = g * 64;
  const int tid = threadIdx.x;
  const int w = tid >> 5;
  const int lane = tid & 31;
  const int l15 = lane & 15;
  const int kh = lane >> 4;

  int rowBase = mBlk * 256 + w * 32;
  if (rowBase > M - 32) rowBase = M - 32;  // wave-uniform clamp; dup tiles write identical data

  const float* pA0 = A + (size_t)(rowBase + l15) * K + (kh * 2);
  const float* pA1 = pA0 + (size_t)16 * K;

  v8f acc[2][4] = {};

  // prologue: stage chunk 0 into buffer 0 (coalesced 64-wide rows)
  {
    const float* Bg = B + nBase;
#pragma unroll
    for (int r = 0; r < (KC * 64) / 256; ++r) {
      int idx = tid + r * 256;
      int kk = idx >> 6, c = idx & 63;
      Bs[0][((kk >> 1) * 64 + c) * 2 + (kk & 1)] = Bg[(size_t)kk * N + c];
    }
  }
  __syncthreads();

  for (int kc = 0; kc < K; kc += KC) {
    const int buf = (kc / KC) & 1;
    // prefetch/stage next chunk into the other buffer while computing this one
    if (kc + KC < K) {
      __builtin_prefetch(pA0 + kc + KC, 0, 1);
      __builtin_prefetch(pA1 + kc + KC, 0, 1);
      const float* Bg = B + (size_t)(kc + KC) * N + nBase;
#pragma unroll
      for (int r = 0; r < (KC * 64) / 256; ++r) {
        int idx = tid + r * 256;
        int kk = idx >> 6, c = idx & 63;
        Bs[buf ^ 1][((kk >> 1) * 64 + c) * 2 + (kk & 1)] = Bg[(size_t)kk * N + c];
      }
    }
    const v2f* Bs2 = (const v2f*)Bs[buf];
#pragma unroll
    for (int t = 0; t < KC / 4; ++t) {
      const int k = kc + t * 4;
      v2f a0 = *(const v2f*)(pA0 + k);
      v2f a1 = *(const v2f*)(pA1 + k);
      const int p = 2 * t + kh;
#pragma unroll
      for (int j = 0; j < 4; ++j) {
        v2f bv = Bs2[p * 64 + j * 16 + l15];
        acc[0][j] = __builtin_amdgcn_wmma_f32_16x16x4_f32(
            false, a0, false, bv, (short)0, acc[0][j], false, false);
        acc[1][j] = __builtin_amdgcn_wmma_f32_16x16x4_f32(
            false, a1, false, bv, (short)0, acc[1][j], false, false);
      }
    }
    __syncthreads();
  }

#pragma unroll
  for (int h = 0; h < 2; ++h) {
#pragma unroll
    for (int j = 0; j < 4; ++j) {
      const int col = nBase + j * 16 + l15;
      const float bb = bias ? bias[col] : 0.0f;
#pragma unroll
      for (int v = 0; v < 8; ++v) {
        const int row = rowBase + h * 16 + kh * 8 + v;
        C[(size_t)row * N + col] = acc[h][j][v] + bb;
      }
    }
  }
}

// ---------------------------------------------------------------------------
// Edge scatter: Agg[dst] += H[src] * dinv[src]*dinv[dst].  One wave per edge,
// lane-strided features -> each global op touches 32 consecutive floats.
// ---------------------------------------------------------------------------
template <int F>
__global__ void scatter_edges(const int* __restrict__ src, const int* __restrict__ dst,
                              const float* __restrict__ dinv,
                              const float* __restrict__ H, float* __restrict__ Agg,
                              int E) {
  const int wave = blockIdx.x * (blockDim.x >> 5) + (threadIdx.x >> 5);
  if (wave >= E) return;
  const int lane = threadIdx.x & 31;
  const int s = src[wave];
  const int d = dst[wave];
  const float norm = dinv[s] * dinv[d];
  const float* hs = H + (size_t)s * F;
  float* ad = Agg + (size_t)d * F;
#pragma unroll
  for (int f = lane; f < F; f += 32)
    unsafeAtomicAdd(&ad[f], hs[f] * norm);
}

// hidden = relu(agg + h1*dinv^2 + b1)   (in-place on agg); F = 2^LOGF
template <int LOGF>
__global__ void finalize_hidden(float* __restrict__ agg, const float* __restrict__ h1,
                                const float* __restrict__ dinv,
                                const float* __restrict__ b, size_t total) {
  size_t idx = (size_t)blockIdx.x * blockDim.x + threadIdx.x;
  if (idx >= total) return;
  int i = (int)(idx >> LOGF);
  int f = (int)(idx & ((1u << LOGF) - 1));
  float di = dinv[i];
  float v = agg[idx] + h1[idx] * di * di + b[f];
  agg[idx] = fmaxf(v, 0.0f);
}

// S += hidden * dinv^2   (fold self-loop into aggregated hidden before GEMM)
template <int LOGF>
__global__ void add_self(float* __restrict__ S, const float* __restrict__ hidden,
                         const float* __restrict__ dinv, size_t total) {
  size_t idx = (size_t)blockIdx.x * blockDim.x + threadIdx.x;
  if (idx >= total) return;
  int i = (int)(idx >> LOGF);
  float di = dinv[i];
  S[idx] += hidden[idx] * di * di;
}

// ---------------------------------------------------------------------------
// Launch. Inputs: x, edge_index, W1, b1, W_mu, b_mu, W_logstd, b_logstd.
// Output: [mu (n*128) | logstd (n*128)].
// Math: hidden = relu(Agg(x@W1) + b1); Agg is linear in features, so
//       mu = (Agg(hidden) + hidden*dinv^2) @ W_mu + b_mu   (same for logstd),
// which halves layer-2 edge passes and fuses bias into the final GEMMs.
// ---------------------------------------------------------------------------
extern "C" void kernel_launch(void* const* d_in, const int* in_sizes, int n_in,
                              void* d_out, int out_size, void* d_ws, size_t ws_size,
                              hipStream_t stream) {
  (void)n_in; (void)out_size; (void)ws_size;
  const float* x   = (const float*)d_in[0];
  const int*   ei  = (const int*)d_in[1];
  const float* W1  = (const float*)d_in[2];
  const float* b1  = (const float*)d_in[3];
  const float* Wmu = (const float*)d_in[4];
  const float* bmu = (const float*)d_in[5];
  const float* Wls = (const float*)d_in[6];
  const float* bls = (const float*)d_in[7];

  const int FIN = 1024, FH = 256, FO = 128;
  const int n = in_sizes[0] / FIN;  // 100000
  const int E = in_sizes[1] / 2;    // 3200000
  const int* src = ei;
  const int* dst = ei + E;

  float* ws   = (float*)d_ws;
  float* dinv = ws;                      // n floats (deg -> rsqrt in place)
  float* h1   = ws + (size_t)(1 << 17);  // n*FH floats (later reused as T)
  float* agg  = h1 + (size_t)n * FH;     // n*FH floats (later = hidden)

  float* outMu = (float*)d_out;
  float* outLs = outMu + (size_t)n * FO;

  const dim3 blk(256);
  auto cdiv = [](long long a, long long b) { return (unsigned)((a + b - 1) / b); };
  const size_t nFH = (size_t)n * FH;

  // --- degree & D^-1/2 ---
  deg_init <<<cdiv(n, 256), blk, 0, stream>>>(dinv, n);
  deg_count<<<cdiv(E, 256), blk, 0, stream>>>(dst, dinv, E);
  deg_rsqrt<<<cdiv(n, 256), blk, 0, stream>>>(dinv, n);

  const unsigned mBlocks = cdiv(n, 256);  // 256 rows per block

  // --- layer 1: h1 = x @ W1 (WMMA fp32, LDS-staged B) ---
  gemm_wmma_f32<<<mBlocks * (FH / 64), blk, 0, stream>>>(x, W1, nullptr, h1, n, FIN, FH);

  // agg = Agg_edges(h1)
  zero_f32<<<cdiv((long long)nFH, 256), blk, 0, stream>>>(agg, nFH);
  scatter_edges<256><<<cdiv(E, 8), blk, 0, stream>>>(src, dst, dinv, h1, agg, E);
  // hidden = relu(agg + h1*dinv^2 + b1)   (in agg)
  finalize_hidden<8><<<cdiv((long long)nFH, 256), blk, 0, stream>>>(agg, h1, dinv, b1, nFH);

  // --- layer 2 (aggregate-then-project): T = Agg(hidden) + hidden*dinv^2 ---
  zero_f32<<<cdiv((long long)nFH, 256), blk, 0, stream>>>(h1, nFH);  // reuse h1 as T
  scatter_edges<256><<<cdiv(E, 8), blk, 0, stream>>>(src, dst, dinv, agg, h1, E);
  add_self<8><<<cdiv((long long)nFH, 256), blk, 0, stream>>>(h1, agg, dinv, nFH);

  // mu = T @ W_mu + b_mu ; logstd = T @ W_logstd + b_logstd
  gemm_wmma_f32<<<mBlocks * (FO / 64), blk, 0, stream>>>(h1, Wmu, bmu, outMu, n, FH, FO);
  gemm_wmma_f32<<<mBlocks * (FO / 64), blk, 0, stream>>>(h1, Wls, bls, outLs, n, FH, FO);
}